// TSPTransformerDecoderLayerFinal_25924422599013
// MI455X (gfx1250) — compile-verified
//
#include <hip/hip_runtime.h>
#include <hip/hip_bf16.h>
#include <math.h>

typedef float v2f __attribute__((ext_vector_type(2)));
typedef float v8f __attribute__((ext_vector_type(8)));

#define B_   256
#define S_   1000
#define D_   256
#define H_   8
#define HD_  32
#define EPS_ 1e-5f
#define CLIP_ 10.0f

// ---------------- block reduction helpers ----------------
__device__ __forceinline__ float block_reduce_sum(float v, float* red) {
    int t = threadIdx.x;
    red[t] = v; __syncthreads();
    for (int off = blockDim.x >> 1; off > 0; off >>= 1) {
        if (t < off) red[t] += red[t + off];
        __syncthreads();
    }
    float r = red[0]; __syncthreads();
    return r;
}
__device__ __forceinline__ float block_reduce_max(float v, float* red) {
    int t = threadIdx.x;
    red[t] = v; __syncthreads();
    for (int off = blockDim.x >> 1; off > 0; off >>= 1) {
        if (t < off) red[t] = fmaxf(red[t], red[t + off]);
        __syncthreads();
    }
    float r = red[0]; __syncthreads();
    return r;
}

// ---------------- K1: self-attn(T=1) + LN1 + fold ca_wk into query ----------------
// grid = B, block = 256
__global__ void k1_prep(const float* __restrict__ ht,
                        const float* __restrict__ sa_wv, const float* __restrict__ sa_bv,
                        const float* __restrict__ sa_wo, const float* __restrict__ sa_bo,
                        const float* __restrict__ n1_g,  const float* __restrict__ n1_b,
                        const float* __restrict__ ca_wq, const float* __restrict__ ca_bq,
                        const float* __restrict__ ca_wk,
                        float* __restrict__ q_out, float* __restrict__ qt_out) {
    __shared__ float x[D_], tmp[D_], red[D_];
    int b = blockIdx.x, t = threadIdx.x;
    x[t] = ht[b * D_ + t];
    __syncthreads();
    // v = h_t @ sa_wv + bv   (softmax over 1 key == 1 -> attn out == v)
    float acc = sa_bv[t];
    for (int e = 0; e < D_; ++e) acc += x[e] * sa_wv[e * D_ + t];
    tmp[t] = acc; __syncthreads();
    // o = v @ sa_wo + bo ; pre = h_t + o
    acc = sa_bo[t];
    for (int e = 0; e < D_; ++e) acc += tmp[e] * sa_wo[e * D_ + t];
    float pre = x[t] + acc;
    // LayerNorm 1
    float m   = block_reduce_sum(pre, red) * (1.0f / D_);
    float dv  = pre - m;
    float var = block_reduce_sum(dv * dv, red) * (1.0f / D_);
    float q   = dv * rsqrtf(var + EPS_) * n1_g[t] + n1_b[t];
    q_out[b * D_ + t] = q;
    x[t] = q; __syncthreads();
    // qc = query @ ca_wq + bq
    acc = ca_bq[t];
    for (int e = 0; e < D_; ++e) acc += x[e] * ca_wq[e * D_ + t];
    tmp[t] = acc; __syncthreads();
    // qtilde[b][h][t] = (1/sqrt(HD)) * sum_hd qc[h*32+hd] * ca_wk[t][h*32+hd]
    float a8[H_];
#pragma unroll
    for (int h = 0; h < H_; ++h) a8[h] = 0.f;
    for (int c = 0; c < D_; ++c) a8[c >> 5] += tmp[c] * ca_wk[t * D_ + c];
    const float rs32 = 0.17677669529663687f; // 1/sqrt(32)
#pragma unroll
    for (int h = 0; h < H_; ++h)  qt_out[(size_t)b * 4096 + h * D_ + t] = a8[h] * rs32;
#pragma unroll
    for (int h = H_; h < 16; ++h) qt_out[(size_t)b * 4096 + h * D_ + t] = 0.f;
}

// ---------------- K2: cross-attn scores via WMMA f32 16x16x4 ----------------
// grid = (63, B), block = 32 (one wave). Tile: 16 s-rows x 16 cols (8 heads valid), K=256.
__global__ void k2_cross_scores(const float* __restrict__ mem,
                                const float* __restrict__ qt,
                                const float* __restrict__ mask,
                                float* __restrict__ sc) {
    int b    = blockIdx.y;
    int s0   = blockIdx.x * 16;
    int lane = threadIdx.x;
    int half = lane >> 4, mrow = lane & 15;
    int srow = s0 + mrow; if (srow > S_ - 1) srow = S_ - 1;   // clamp, keep EXEC all-ones
    const float* aptr = mem + (size_t)b * (S_ * D_) + (size_t)srow * D_ + half * 2;
    const float* bptr = qt  + (size_t)b * 4096 + mrow * D_ + half * 2;
    v8f c = {};
    for (int k = 0; k < D_; k += 4) {
        v2f a  = *(const v2f*)(aptr + k);
        v2f bb = *(const v2f*)(bptr + k);
        c = __builtin_amdgcn_wmma_f32_16x16x4_f32(false, a, false, bb, (short)0, c, false, false);
    }
    int h = mrow;
    if (h < H_) {
#pragma unroll
        for (int v = 0; v < 8; ++v) {
            int s = s0 + v + half * 8;
            if (s < S_)
                sc[(size_t)b * (H_ * S_) + (size_t)h * S_ + s] = c[v] + mask[b * S_ + s];
        }
    }
}

// ---------------- K3: softmax over S per (b,h) ----------------
// grid = B*H, block = 256
__global__ void k3_softmax(float* __restrict__ sc) {
    __shared__ float red[256];
    size_t base = (size_t)blockIdx.x * S_;
    int t = threadIdx.x;
    float m = -3.4e38f;
    for (int s = t; s < S_; s += 256) m = fmaxf(m, sc[base + s]);
    m = block_reduce_max(m, red);
    float sum = 0.f;
    for (int s = t; s < S_; s += 256) sum += expf(sc[base + s] - m);
    sum = block_reduce_sum(sum, red);
    float inv = 1.0f / sum;
    for (int s = t; s < S_; s += 256) sc[base + s] = expf(sc[base + s] - m) * inv;
}

// ---------------- K4: u[b][h][e] = sum_s a[b][h][s] * mem[b][s][e] ----------------
// grid = B, block = 256 (thread == column e); probs broadcast through LDS
__global__ void k4_av(const float* __restrict__ mem, const float* __restrict__ p,
                      float* __restrict__ u) {
    __shared__ float aa[H_][32];
    int b = blockIdx.x, t = threadIdx.x;
    float acc[H_];
#pragma unroll
    for (int h = 0; h < H_; ++h) acc[h] = 0.f;
    for (int c0 = 0; c0 < S_; c0 += 32) {
        int h = t >> 5, si = t & 31;
        int s = c0 + si;
        aa[h][si] = (s < S_) ? p[(size_t)b * (H_ * S_) + (size_t)h * S_ + s] : 0.f;
        __syncthreads();
        int lim = (S_ - c0 < 32) ? (S_ - c0) : 32;
        for (int i = 0; i < lim; ++i) {
            float mv = mem[(size_t)b * (S_ * D_) + (size_t)(c0 + i) * D_ + t];
#pragma unroll
            for (int h2 = 0; h2 < H_; ++h2) acc[h2] += aa[h2][i] * mv;
        }
        __syncthreads();
    }
#pragma unroll
    for (int h = 0; h < H_; ++h) u[(size_t)b * (H_ * D_) + h * D_ + t] = acc[h];
}

// ---------------- K5: attn-out proj + LN2 + fold p_wk into pointer query ----------------
// grid = B, block = 256
__global__ void k5_mid(const float* __restrict__ u,
                       const float* __restrict__ ca_wv, const float* __restrict__ ca_bv,
                       const float* __restrict__ ca_wo, const float* __restrict__ ca_bo,
                       const float* __restrict__ query,
                       const float* __restrict__ n2_g, const float* __restrict__ n2_b,
                       const float* __restrict__ p_wq, const float* __restrict__ p_bq,
                       const float* __restrict__ p_wk, const float* __restrict__ p_bk,
                       float* __restrict__ qpt, float* __restrict__ bias) {
    __shared__ float uu[H_ * D_];
    __shared__ float o[D_], tmp[D_], red[D_];
    int b = blockIdx.x, t = threadIdx.x;
#pragma unroll
    for (int i = 0; i < H_; ++i) uu[i * D_ + t] = u[(size_t)b * (H_ * D_) + i * D_ + t];
    __syncthreads();
    // o[d] = sum_e u[h(d)][e]*ca_wv[e][d] + bv[d]   (sum_s a == 1 -> bias passes through)
    int h = t >> 5;
    float acc = ca_bv[t];
    for (int e = 0; e < D_; ++e) acc += uu[h * D_ + e] * ca_wv[e * D_ + t];
    o[t] = acc; __syncthreads();
    // cross = o @ ca_wo + bo ; pre = query + cross
    acc = ca_bo[t];
    for (int e = 0; e < D_; ++e) acc += o[e] * ca_wo[e * D_ + t];
    float pre = query[b * D_ + t] + acc;
    float m   = block_reduce_sum(pre, red) * (1.0f / D_);
    float dv  = pre - m;
    float var = block_reduce_sum(dv * dv, red) * (1.0f / D_);
    float hq  = dv * rsqrtf(var + EPS_) * n2_g[t] + n2_b[t];
    tmp[t] = hq; __syncthreads();
    // qp = (h_q @ p_wq + p_bq) / sqrt(D)
    acc = p_bq[t];
    for (int e = 0; e < D_; ++e) acc += tmp[e] * p_wq[e * D_ + t];
    float qp = acc * 0.0625f;
    o[t] = qp; __syncthreads();
    // qp_tilde[b][j][t] = sum_hd qp[j*32+hd] * p_wk[t][j*32+hd]
    float a8[H_];
#pragma unroll
    for (int j = 0; j < H_; ++j) a8[j] = 0.f;
    for (int c = 0; c < D_; ++c) a8[c >> 5] += o[c] * p_wk[t * D_ + c];
#pragma unroll
    for (int j = 0; j < H_; ++j)  qpt[(size_t)b * 4096 + j * D_ + t] = a8[j];
#pragma unroll
    for (int j = H_; j < 16; ++j) qpt[(size_t)b * 4096 + j * D_ + t] = 0.f;
    if (t < H_) {
        float bacc = 0.f;
        for (int hd = 0; hd < HD_; ++hd) bacc += o[t * HD_ + hd] * p_bk[t * HD_ + hd];
        bias[b * H_ + t] = bacc;   // tanh is nonlinear: bias must NOT be dropped
    }
}

// ---------------- K6: pointer logits via WMMA (torch-view index algebra folded in) ----------------
// attn[bb*8+j, s2] = memflat[s2*65536 + bb*256 + :256] . qpt[bb*8+j,:] + bias + mask
// grid = (63, B), block = 32
__global__ void k6_ptr_scores(const float* __restrict__ mem,
                              const float* __restrict__ qpt,
                              const float* __restrict__ bias,
                              const float* __restrict__ mask,
                              float* __restrict__ pt) {
    int bb   = blockIdx.y;
    int s0   = blockIdx.x * 16;
    int lane = threadIdx.x;
    int half = lane >> 4, mrow = lane & 15;
    int srow = s0 + mrow; if (srow > S_ - 1) srow = S_ - 1;
    const float* aptr = mem + (size_t)srow * (B_ * D_) + (size_t)bb * D_ + half * 2;
    const float* bptr = qpt + (size_t)bb * 4096 + mrow * D_ + half * 2;
    v8f c = {};
    for (int k = 0; k < D_; k += 4) {
        v2f a  = *(const v2f*)(aptr + k);
        v2f bv = *(const v2f*)(bptr + k);
        c = __builtin_amdgcn_wmma_f32_16x16x4_f32(false, a, false, bv, (short)0, c, false, false);
    }
    int j = mrow;
    if (j < H_) {
        float bj = bias[bb * H_ + j];
#pragma unroll
        for (int v = 0; v < 8; ++v) {
            int s = s0 + v + half * 8;
            if (s < S_) {
                float att = c[v] + bj + mask[bb * S_ + s];
                pt[(size_t)bb * (H_ * S_) + (size_t)j * S_ + s] = CLIP_ * tanhf(att);
            }
        }
    }
}

// ---------------- K7: per-head softmax over S, then mean over heads ----------------
// grid = B, block = 256
__global__ void k7_out(const float* __restrict__ pt, float* __restrict__ out) {
    __shared__ float red[256];
    __shared__ float mj[H_], lj[H_];
    int bb = blockIdx.x, t = threadIdx.x;
    for (int j = 0; j < H_; ++j) {
        size_t base = (size_t)bb * (H_ * S_) + (size_t)j * S_;
        float m = -3.4e38f;
        for (int s = t; s < S_; s += 256) m = fmaxf(m, pt[base + s]);
        m = block_reduce_max(m, red);
        float sum = 0.f;
        for (int s = t; s < S_; s += 256) sum += expf(pt[base + s] - m);
        sum = block_reduce_sum(sum, red);
        if (t == 0) { mj[j] = m; lj[j] = 1.0f / sum; }
        __syncthreads();
    }
    for (int s = t; s < S_; s += 256) {
        float acc = 0.f;
#pragma unroll
        for (int j = 0; j < H_; ++j)
            acc += expf(pt[(size_t)bb * (H_ * S_) + (size_t)j * S_ + s] - mj[j]) * lj[j];
        out[(size_t)bb * S_ + s] = acc * 0.125f;
    }
}

extern "C" void kernel_launch(void* const* d_in, const int* in_sizes, int n_in,
                              void* d_out, int out_size, void* d_ws, size_t ws_size,
                              hipStream_t stream) {
    const float* ht     = (const float*)d_in[0];
    const float* mem    = (const float*)d_in[1];
    const float* mask   = (const float*)d_in[2];
    // d_in[3..6] = sa_wq, sa_bq, sa_wk, sa_bk : dead (T=1 -> softmax==1)
    const float* sa_wv  = (const float*)d_in[7];
    const float* sa_bv  = (const float*)d_in[8];
    const float* sa_wo  = (const float*)d_in[9];
    const float* sa_bo  = (const float*)d_in[10];
    const float* ca_wq  = (const float*)d_in[11];
    const float* ca_bq  = (const float*)d_in[12];
    const float* ca_wk  = (const float*)d_in[13];
    // d_in[14] = ca_bk : constant over keys -> cancels in softmax
    const float* ca_wv  = (const float*)d_in[15];
    const float* ca_bv  = (const float*)d_in[16];
    const float* ca_wo  = (const float*)d_in[17];
    const float* ca_bo  = (const float*)d_in[18];
    const float* n1_g   = (const float*)d_in[19];
    const float* n1_b   = (const float*)d_in[20];
    const float* n2_g   = (const float*)d_in[21];
    const float* n2_b   = (const float*)d_in[22];
    const float* p_wq   = (const float*)d_in[23];
    const float* p_bq   = (const float*)d_in[24];
    const float* p_wk   = (const float*)d_in[25];
    const float* p_bk   = (const float*)d_in[26];
    float* out = (float*)d_out;

    // workspace layout (floats)
    float* ws       = (float*)d_ws;
    float* w_query  = ws;                         // B*D               = 65536
    float* w_qt     = w_query + (size_t)B_ * D_;  // B*16*D            = 1048576
    float* w_sc     = w_qt + (size_t)B_ * 16 * D_;// B*H*S             = 2048000 (reused as pointer logits)
    float* w_u      = w_sc + (size_t)B_ * H_ * S_;// B*H*D             = 524288
    float* w_qpt    = w_u + (size_t)B_ * H_ * D_; // B*16*D            = 1048576
    float* w_bias   = w_qpt + (size_t)B_ * 16 * D_;// B*H              = 2048
    float* w_pt     = w_sc;                       // reuse score buffer for pointer logits

    dim3 tiles(63, B_);   // ceil(1000/16) x batches

    k1_prep<<<B_, 256, 0, stream>>>(ht, sa_wv, sa_bv, sa_wo, sa_bo, n1_g, n1_b,
                                    ca_wq, ca_bq, ca_wk, w_query, w_qt);
    k2_cross_scores<<<tiles, 32, 0, stream>>>(mem, w_qt, mask, w_sc);
    k3_softmax<<<B_ * H_, 256, 0, stream>>>(w_sc);
    k4_av<<<B_, 256, 0, stream>>>(mem, w_sc, w_u);
    k5_mid<<<B_, 256, 0, stream>>>(w_u, ca_wv, ca_bv, ca_wo, ca_bo, w_query,
                                   n2_g, n2_b, p_wq, p_bq, p_wk, p_bk, w_qpt, w_bias);
    k6_ptr_scores<<<tiles, 32, 0, stream>>>(mem, w_qpt, w_bias, mask, w_pt);
    k7_out<<<B_, 256, 0, stream>>>(w_pt, out);
}